// PermutohedralLayer_75574244540745
// MI455X (gfx1250) — compile-verified
//
#include <hip/hip_runtime.h>
#include <stdint.h>

// ---------------- problem constants (from reference) ----------------
constexpr int kB = 2, kC = 21, kH = 256, kW = 256;
constexpr int kN = kB * kH * kW;              // 131072 points
constexpr int kR = 6;                          // d+1 simplex vertices (d = 5)
constexpr unsigned kP = 1u << 21;              // hash capacity (load <= 0.375)
constexpr int kMaxU = kN * kR;                 // upper bound on unique lattice pts
constexpr unsigned long long kEmpty = 0xFFFFFFFFFFFFFFFFull;

// ---------------- TDM: 1-D tile (4-byte elems) global -> LDS ----------------
#if defined(__gfx1250__) && __has_builtin(__builtin_amdgcn_tensor_load_to_lds)
#define USE_TDM 1
typedef unsigned int u32x4 __attribute__((ext_vector_type(4)));
typedef int i32x8 __attribute__((ext_vector_type(8)));
typedef int i32x4 __attribute__((ext_vector_type(4)));

// 6-arg builtin form (amdgpu-toolchain clang-23): (g0, g1, g2, g3, g4, cpol)
__device__ __forceinline__ void tdm_load_1d(unsigned ldsByteAddr, const void* gptr, unsigned nElems) {
  unsigned long long ga = (unsigned long long)(uintptr_t)gptr;
  u32x4 g0;
  g0.x = 1u;                                            // count=1, user descriptor
  g0.y = ldsByteAddr;                                   // LDS dest (bytes)
  g0.z = (unsigned)(ga & 0xFFFFFFFFu);                  // global_addr[31:0]
  g0.w = (unsigned)((ga >> 32) & 0x1FFFFFFu) | (2u << 30); // global_addr[56:32], type=2
  i32x8 g1;
  g1[0] = (int)(2u << 16);                              // data_size = 4B
  g1[1] = (int)((nElems & 0xFFFFu) << 16);              // tensor_dim0 lo16 @ bits 63:48
  g1[2] = (int)((nElems >> 16) | (1u << 16));           // tensor_dim0 hi16, tensor_dim1 = 1
  g1[3] = (int)((nElems & 0xFFFFu) << 16);              // tile_dim0 @ bits 127:112
  g1[4] = 1;                                            // tile_dim1 = 1, tile_dim2 = 0
  g1[5] = (int)nElems;                                  // tensor_dim0_stride lo32
  g1[6] = 0;
  g1[7] = 0;
  i32x4 z4 = {0, 0, 0, 0};
  i32x8 z8 = {0, 0, 0, 0, 0, 0, 0, 0};
  __builtin_amdgcn_tensor_load_to_lds(g0, g1, z4, z4, z8, 0);
}
#endif

// Stage a block's contiguous idx/bary tiles (256*6 each) into LDS.
__device__ __forceinline__ void stage_tiles(const int* __restrict__ idx,
                                            const float* __restrict__ bary,
                                            int nBase, int* sIdx, float* sBary) {
  const int tile = 256 * kR;                            // 1536 elements
#ifdef USE_TDM
  if (threadIdx.x < 32u) {                              // wave 0 drives the TDM
    tdm_load_1d((unsigned)(uintptr_t)sIdx,  idx  + (size_t)nBase * kR, tile);
    tdm_load_1d((unsigned)(uintptr_t)sBary, bary + (size_t)nBase * kR, tile);
#if __has_builtin(__builtin_amdgcn_s_wait_tensorcnt)
    __builtin_amdgcn_s_wait_tensorcnt(0);
#endif
  }
  __syncthreads();
#else
  for (int t = threadIdx.x; t < tile; t += 256) {
    sIdx[t]  = idx[(size_t)nBase * kR + t];
    sBary[t] = bary[(size_t)nBase * kR + t];
  }
  __syncthreads();
#endif
}

// key packing: coord_i + 2048 in 12-bit field i (i<5), batch id at bit 60
__device__ __forceinline__ unsigned long long mix64(unsigned long long k) {
  k ^= k >> 33; k *= 0xff51afd7ed558ccdull;
  k ^= k >> 33; k *= 0xc4ceb9fe1a85ec53ull;
  k ^= k >> 33; return k;
}

__device__ __forceinline__ void hash_insert(unsigned long long* hk, unsigned long long key) {
  unsigned h = (unsigned)mix64(key) & (kP - 1);
  for (;;) {
    unsigned long long prev = atomicCAS(&hk[h], kEmpty, key);
    if (prev == kEmpty || prev == key) return;
    h = (h + 1) & (kP - 1);
  }
}

__device__ __forceinline__ int hash_find(const unsigned long long* hk, unsigned long long key) {
  unsigned h = (unsigned)mix64(key) & (kP - 1);
  for (;;) {
    unsigned long long k = hk[h];
    if (k == key) return (int)h;
    if (k == kEmpty) return -1;
    h = (h + 1) & (kP - 1);
  }
}

// ---------------- 1) per-point: elevate, simplex, keys, insert ----------------
__global__ __launch_bounds__(256)
void k_point(const float* __restrict__ img, unsigned long long* __restrict__ hkey,
             unsigned long long* __restrict__ keys, float* __restrict__ bary) {
  int n = blockIdx.x * blockDim.x + threadIdx.x;
  if (n >= kN) return;
  int b = n >> 16, hw = n & 0xFFFF;
  int h = hw >> 8, w = hw & 255;

  // features: [x/80, y/80, r/13, g/13, b/13]
  float f[5];
  f[0] = (float)w * (1.0f / 80.0f);
  f[1] = (float)h * (1.0f / 80.0f);
  const int ib = (b * 3) << 16;
  f[2] = img[ib + hw]             * (1.0f / 13.0f);
  f[3] = img[ib + (1 << 16) + hw] * (1.0f / 13.0f);
  f[4] = img[ib + (2 << 16) + hw] * (1.0f / 13.0f);

  // scale_i = sqrt(2/3)*6 / sqrt((i+1)(i+2))
  const float sc[5] = {3.46410161514f, 2.0f, 1.41421356237f, 1.09544511501f, 0.894427191f};
  float g[5];
#pragma unroll
  for (int i = 0; i < 5; i++) g[i] = f[i] * sc[i];

  // elevated[j] = sum_{i>=j} g[i] - j*g[j-1]  (suffix-sum form of E)
  float el[6];
  float sm = 0.f;
  el[5] = -5.f * g[4];
  sm = g[4];        el[4] = sm - 4.f * g[3];
  sm += g[3];       el[3] = sm - 3.f * g[2];
  sm += g[2];       el[2] = sm - 2.f * g[1];
  sm += g[1];       el[1] = sm - 1.f * g[0];
  sm += g[0];       el[0] = sm;

  float rem0[6], di[6], rsum = 0.f;
#pragma unroll
  for (int i = 0; i < 6; i++) {
    float v  = el[i] * (1.0f / 6.0f);
    float up = ceilf(v) * 6.f, dn = floorf(v) * 6.f;
    float r0 = (up - el[i] < el[i] - dn) ? up : dn;
    rem0[i] = r0; di[i] = el[i] - r0; rsum += r0;
  }
  int ssum = (int)rintf(rsum * (1.0f / 6.0f));

  int rank[6];
#pragma unroll
  for (int i = 0; i < 6; i++) {
    int rk = 0;
#pragma unroll
    for (int k = 0; k < 6; k++) {
      if (k == i) continue;
      rk += (k > i) ? (di[i] < di[k]) : (di[i] <= di[k]);
    }
    rank[i] = rk + ssum;
  }
#pragma unroll
  for (int i = 0; i < 6; i++) {        // wrap (matches reference where-clauses)
    if (rank[i] < 0)      { rank[i] += 6; rem0[i] += 6.f; }
    else if (rank[i] > 5) { rank[i] -= 6; rem0[i] -= 6.f; }
  }

  // barycentric weights
  float bz[7] = {0.f, 0.f, 0.f, 0.f, 0.f, 0.f, 0.f};
#pragma unroll
  for (int i = 0; i < 6; i++) {
    float t = (el[i] - rem0[i]) * (1.0f / 6.0f);
    bz[5 - rank[i]] += t;
    bz[6 - rank[i]] -= t;
  }
  bz[0] += 1.0f + bz[6];

  // pack 6 vertex keys and insert into hash
  const unsigned long long pw[5] = {1ull, 4096ull, 16777216ull, 68719476736ull, 281474976710656ull};
  const unsigned long long base = (unsigned long long)b << 60;
  int c0[5];
#pragma unroll
  for (int i = 0; i < 5; i++) c0[i] = (int)rintf(rem0[i]);
#pragma unroll
  for (int r = 0; r < 6; r++) {
    unsigned long long kk = base;
#pragma unroll
    for (int i = 0; i < 5; i++) {
      int ci = c0[i] + r - ((rank[i] > 5 - r) ? 6 : 0);
      kk += (unsigned long long)(ci + 2048) * pw[i];
    }
    keys[n * kR + r] = kk;
    bary[n * kR + r] = bz[r];
    hash_insert(hkey, kk);
  }
}

// ---------------- 2) compact hash slots -> dense ids ----------------
__global__ __launch_bounds__(256)
void k_assign(const unsigned long long* __restrict__ hkey, int* __restrict__ hid,
              unsigned long long* __restrict__ ukey, int* __restrict__ cnt) {
  int s = blockIdx.x * blockDim.x + threadIdx.x;
  if (s >= (int)kP) return;
  unsigned long long k = hkey[s];
  if (k == kEmpty) return;
  int id = atomicAdd(cnt, 1);
  hid[s] = id;
  ukey[id] = k;
}

// ---------------- 3) per-vertex key -> dense id ----------------
__global__ __launch_bounds__(256)
void k_lookup(const unsigned long long* __restrict__ keys,
              const unsigned long long* __restrict__ hkey,
              const int* __restrict__ hid, int* __restrict__ idx) {
  int m = blockIdx.x * blockDim.x + threadIdx.x;
  if (m >= kMaxU) return;
  idx[m] = hid[hash_find(hkey, keys[m])];
}

// ---------------- 4) splat (TDM-staged tiles; scatter-add into L2 lattice) ----------------
__global__ __launch_bounds__(256)
void k_splat(const float* __restrict__ x, const int* __restrict__ idx,
             const float* __restrict__ bary, float* __restrict__ lat) {
  __shared__ int   sIdx[256 * kR];
  __shared__ float sBary[256 * kR];
  const int nBase = blockIdx.x * 256;
  stage_tiles(idx, bary, nBase, sIdx, sBary);

  const int n = nBase + threadIdx.x;
  const int li = threadIdx.x * kR;
  int b = n >> 16, hw = n & 0xFFFF;
  int i6[6]; float w6[6];
#pragma unroll
  for (int r = 0; r < 6; r++) { i6[r] = sIdx[li + r]; w6[r] = sBary[li + r]; }
  const float* xb = x + (((size_t)b * kC) << 16) + hw;
  for (int c = 0; c < kC; c++) {
    float v = xb[(size_t)c << 16];
#pragma unroll
    for (int r = 0; r < 6; r++)
      atomicAdd(&lat[(size_t)i6[r] * kC + c], w6[r] * v);
  }
}

// ---------------- 5) blur along one lattice direction ----------------
__global__ __launch_bounds__(256)
void k_blur(const unsigned long long* __restrict__ ukey,
            const unsigned long long* __restrict__ hkey,
            const int* __restrict__ hid, const int* __restrict__ cnt,
            const float* __restrict__ latIn, float* __restrict__ latOut,
            long long delta) {
  int id = blockIdx.x * blockDim.x + threadIdx.x;
  if (id >= *cnt) return;
  unsigned long long key = ukey[id];
  int s1 = hash_find(hkey, (unsigned long long)((long long)key + delta));
  int s2 = hash_find(hkey, (unsigned long long)((long long)key - delta));
  int id1 = (s1 >= 0) ? hid[s1] : -1;
  int id2 = (s2 >= 0) ? hid[s2] : -1;
  const float* a  = latIn + (size_t)id * kC;
  const float* p1 = (id1 >= 0) ? latIn + (size_t)id1 * kC : nullptr;
  const float* p2 = (id2 >= 0) ? latIn + (size_t)id2 * kC : nullptr;
  float* o = latOut + (size_t)id * kC;
  for (int c = 0; c < kC; c++) {
    float u1 = p1 ? p1[c] : 0.f;
    float u2 = p2 ? p2[c] : 0.f;
    o[c] = a[c] + 0.5f * (u1 + u2);
  }
}

// ---------------- 6) slice (TDM-staged tiles; gather + weight) ----------------
__global__ __launch_bounds__(256)
void k_slice(const int* __restrict__ idx, const float* __restrict__ bary,
             const float* __restrict__ lat, float* __restrict__ out) {
  __shared__ int   sIdx[256 * kR];
  __shared__ float sBary[256 * kR];
  const int nBase = blockIdx.x * 256;
  stage_tiles(idx, bary, nBase, sIdx, sBary);

  const int n = nBase + threadIdx.x;
  const int li = threadIdx.x * kR;
  int i6[6]; float w6[6];
#pragma unroll
  for (int r = 0; r < 6; r++) { i6[r] = sIdx[li + r]; w6[r] = sBary[li + r]; }
#pragma unroll
  for (int r = 0; r < 6; r++) __builtin_prefetch(lat + (size_t)i6[r] * kC, 0, 0);

  const int b = n >> 16, hw = n & 0xFFFF;
  const float alpha = 32.0f / 33.0f;                    // 1/(1+2^-5)
  float* ob = out + (((size_t)b * kC) << 16) + hw;
  for (int c = 0; c < kC; c++) {
    float acc = 0.f;
#pragma unroll
    for (int r = 0; r < 6; r++) acc += w6[r] * lat[(size_t)i6[r] * kC + c];
    ob[(size_t)c << 16] = alpha * acc;
  }
}

// ---------------- host-side orchestration ----------------
extern "C" void kernel_launch(void* const* d_in, const int* in_sizes, int n_in,
                              void* d_out, int out_size, void* d_ws, size_t ws_size,
                              hipStream_t stream) {
  const float* x   = (const float*)d_in[0];   // (2,21,256,256)
  const float* img = (const float*)d_in[1];   // (2,3,256,256)
  float* out = (float*)d_out;                 // (2,21,256,256)
  char* ws = (char*)d_ws;

  size_t off = 0;
  auto take = [&](size_t bytes) { size_t o = off; off = (off + bytes + 255) & ~(size_t)255; return o; };
  const size_t oKeys = take((size_t)kMaxU * 8);
  const size_t oBary = take((size_t)kMaxU * 4);
  const size_t oIdx  = take((size_t)kMaxU * 4);
  const size_t oHKey = take((size_t)kP * 8);
  const size_t oHId  = take((size_t)kP * 4);
  const size_t oCnt  = take(256);
  const size_t oUKey = take((size_t)kMaxU * 8);
  const size_t oLatA = take((size_t)kMaxU * kC * 4);
  const size_t oLatB = take((size_t)kMaxU * kC * 4);
  if (off > ws_size) return;                  // workspace too small: bail

  unsigned long long* keys = (unsigned long long*)(ws + oKeys);
  float*              bary = (float*)(ws + oBary);
  int*                idx  = (int*)(ws + oIdx);
  unsigned long long* hkey = (unsigned long long*)(ws + oHKey);
  int*                hid  = (int*)(ws + oHId);
  int*                cnt  = (int*)(ws + oCnt);
  unsigned long long* ukey = (unsigned long long*)(ws + oUKey);
  float*              latA = (float*)(ws + oLatA);
  float*              latB = (float*)(ws + oLatB);

  (void)hipMemsetAsync(hkey, 0xFF, (size_t)kP * 8, stream);
  (void)hipMemsetAsync(cnt, 0, 4, stream);
  (void)hipMemsetAsync(latA, 0, (size_t)kMaxU * kC * 4, stream);

  k_point <<<kN / 256,          256, 0, stream>>>(img, hkey, keys, bary);
  k_assign<<<(int)(kP / 256),   256, 0, stream>>>(hkey, hid, ukey, cnt);
  k_lookup<<<kMaxU / 256,       256, 0, stream>>>(keys, hkey, hid, idx);
  k_splat <<<kN / 256,          256, 0, stream>>>(x, idx, bary, latA);

  // packed-key deltas per blur direction: off = 1 everywhere, -5 at j (first 5 coords packed)
  const long long pw[5] = {1LL, 4096LL, 16777216LL, 68719476736LL, 281474976710656LL};
  long long S = 0; for (int i = 0; i < 5; i++) S += pw[i];
  float* bufs[2] = {latA, latB};
  for (int j = 0; j < 6; j++) {
    long long delta = (j < 5) ? (S - 6 * pw[j]) : S;
    k_blur<<<kMaxU / 256, 256, 0, stream>>>(ukey, hkey, hid, cnt,
                                            bufs[j & 1], bufs[(j + 1) & 1], delta);
  }
  // after 6 ping-pong blurs the result is back in bufs[0] (= latA)
  k_slice<<<kN / 256, 256, 0, stream>>>(idx, bary, bufs[0], out);
}